// Verse_28982439314080
// MI455X (gfx1250) — compile-verified
//
#include <hip/hip_runtime.h>
#include <hip/hip_bf16.h>
#include <stdint.h>

// Verse / node2vec-style NCE SGD update.
//  W      : [500000, 128] f32   (d_in[0])
//  lut    : [1202] f32 sigmoid table (d_in[1])
//  u      : [B] i32             (d_in[2])
//  v_pos  : [B] i32             (d_in[3])
//  v_neg  : [B, 5] i32          (d_in[4])
//  out    : updated W copy      (d_out)
//
// One wave32 per source node i: lane l owns dims [4l, 4l+4).
// du accumulated over the 6 pairs -> single atomic row update for u.

#define EMB_DIM      128
#define NEGK         5
#define LUT_N        1202
#define SGD_LR       0.025f
#define NCE_BIAS     13.122363377404329f   /* ln(500000)        */
#define NCE_NEG_BIAS 11.512925464970229f   /* ln(500000/5)      */

typedef __attribute__((ext_vector_type(4))) unsigned int v4u;
typedef __attribute__((ext_vector_type(8))) int          v8i;
typedef __attribute__((ext_vector_type(4))) int          v4i;

// --- Tensor Data Mover: DMA the 1202-float LUT from global into LDS. ---
// Called from wave 0 only (TDM instruction issues per-wave, EXEC ignored).
__device__ __forceinline__ void lut_load_tdm(const float* __restrict__ gsrc,
                                             float* lds_dst) {
#if __has_builtin(__builtin_amdgcn_tensor_load_to_lds) && \
    __has_builtin(__builtin_amdgcn_s_wait_tensorcnt)
  // Flat pointers to LDS carry the LDS byte offset in addr[31:0] (ISA 10.2).
  uint32_t lds_off = (uint32_t)(uintptr_t)lds_dst;
  uint64_t ga      = (uint64_t)(uintptr_t)gsrc;

  // D# group 0 (ISA 08 §8.3): count=1 | lds_addr | global_addr[56:0] | type=2
  v4u g0;
  g0.x = 1u;                                                   // count = 1 (valid user D#)
  g0.y = lds_off;                                              // lds_addr
  g0.z = (uint32_t)ga;                                         // global_addr[31:0]
  g0.w = (uint32_t)((ga >> 32) & 0x01FFFFFFu) | (2u << 30);    // global_addr[56:32] | type=2

  // D# group 1 (ISA 08 §8.4): data_size=4B, tensor_dim0=tile_dim0=1202 (1-D tile)
  v8i g1;
  g1[0] = 0x20000;                          // [17:16] data_size = 2 -> 4 bytes
  g1[1] = (int)((LUT_N & 0xFFFF) << 16);    // tensor_dim0[15:0]  @ bits 63:48
  g1[2] = (int)(1u << 16);                  // tensor_dim0 hi = 0; tensor_dim1 = 1
  g1[3] = (int)((LUT_N & 0xFFFF) << 16);    // tile_dim0 = 1202   @ bits 127:112
  g1[4] = 0;                                // tile_dim1 = 0 (unused), tile_dim2 = 0
  g1[5] = LUT_N;                            // tensor_dim0_stride[31:0]
  g1[6] = 0;
  g1[7] = 0;

  v4i gz4 = {0, 0, 0, 0};                   // groups 2/3 unused (<=2D tensor)
  v8i gz8 = {0, 0, 0, 0, 0, 0, 0, 0};      // unused trailing descriptor group
  // amdgpu-toolchain (clang-23) 6-arg form:
  // (uint32x4 g0, int32x8 g1, int32x4, int32x4, int32x8, i32 cpol)
  __builtin_amdgcn_tensor_load_to_lds(g0, g1, gz4, gz4, gz8, /*cpol=*/0);
  __builtin_amdgcn_s_wait_tensorcnt(0);     // s_wait_tensorcnt 0
#else
  for (int j = (int)(threadIdx.x & 31u); j < LUT_N; j += 32)
    lds_dst[j] = gsrc[j];
#endif
}

// One dimension of one (u,v) pair: elementwise score -> LUT sigmoid -> grads.
__device__ __forceinline__ float pair_dim(float a, float b, float label, float bias,
                                          const float* lut, float& duacc) {
  float s = a * b - bias;
  s = fminf(6.0f, fmaxf(-6.0f, s));                  // clip(score, -6, 6)
  int t = (int)floorf((s + 6.01f) / 0.01f);          // fast_sigmoid table index
  t = (t < 0) ? 0 : ((t > LUT_N - 1) ? (LUT_N - 1) : t);
  float g  = (label - lut[t]) * SGD_LR;              // (label - sig) * lr
  float du = b * g;                                  // delta for row u
  duacc += du;
  return (a + du) * g;                               // dv = (Wu + du) * g
}

__global__ __launch_bounds__(256, 1) void verse_update_kernel(
    const float* __restrict__ W, const float* __restrict__ lut_g,
    const int* __restrict__ u, const int* __restrict__ v_pos,
    const int* __restrict__ v_neg, float* __restrict__ Wout, int B) {
  __shared__ float lut[LUT_N];

  if ((threadIdx.x >> 5) == 0) {            // wave-uniform guard: only wave 0 issues TDM
    lut_load_tdm(lut_g, lut);
  }
  __syncthreads();

  const int wave = threadIdx.x >> 5;
  const int lane = (int)(threadIdx.x & 31u);
  const int i    = blockIdx.x * ((int)blockDim.x >> 5) + wave;  // source node index
  if (i >= B) return;

  const int ui = u[i];
  const int d0 = lane * 4;                  // this lane's 4 dims
  const float4 wu = *(const float4*)(W + (size_t)ui * EMB_DIM + d0);

  int vlist[1 + NEGK];
  vlist[0] = v_pos[i];
#pragma unroll
  for (int k = 0; k < NEGK; ++k) vlist[k + 1] = v_neg[i * NEGK + k];

  // Prefetch the gathered rows we are about to touch (global_prefetch_b8).
#pragma unroll
  for (int k = 1; k < 1 + NEGK; ++k)
    __builtin_prefetch(W + (size_t)vlist[k] * EMB_DIM + d0, 0, 3);

  float4 duacc = make_float4(0.f, 0.f, 0.f, 0.f);

#pragma unroll
  for (int p = 0; p < 1 + NEGK; ++p) {
    const float label = (p == 0) ? 1.0f : 0.0f;
    const float bias  = (p == 0) ? NCE_BIAS : NCE_NEG_BIAS;
    const size_t voff = (size_t)vlist[p] * EMB_DIM + d0;
    const float4 wv = *(const float4*)(W + voff);   // global_load_b128

    float4 dv;
    dv.x = pair_dim(wu.x, wv.x, label, bias, lut, duacc.x);
    dv.y = pair_dim(wu.y, wv.y, label, bias, lut, duacc.y);
    dv.z = pair_dim(wu.z, wv.z, label, bias, lut, duacc.z);
    dv.w = pair_dim(wu.w, wv.w, label, bias, lut, duacc.w);

    float* o = Wout + voff;                 // hardware f32 atomics (global_atomic_add_f32)
    unsafeAtomicAdd(o + 0, dv.x);
    unsafeAtomicAdd(o + 1, dv.y);
    unsafeAtomicAdd(o + 2, dv.z);
    unsafeAtomicAdd(o + 3, dv.w);
  }

  // Single accumulated update for the u row (6 pairs folded into one atomic set).
  float* ou = Wout + (size_t)ui * EMB_DIM + d0;
  unsafeAtomicAdd(ou + 0, duacc.x);
  unsafeAtomicAdd(ou + 1, duacc.y);
  unsafeAtomicAdd(ou + 2, duacc.z);
  unsafeAtomicAdd(ou + 3, duacc.w);
}

extern "C" void kernel_launch(void* const* d_in, const int* in_sizes, int n_in,
                              void* d_out, int out_size, void* d_ws, size_t ws_size,
                              hipStream_t stream) {
  (void)n_in; (void)d_ws; (void)ws_size;
  const float* W     = (const float*)d_in[0];
  const float* lut   = (const float*)d_in[1];
  const int*   u     = (const int*)d_in[2];
  const int*   v_pos = (const int*)d_in[3];
  const int*   v_neg = (const int*)d_in[4];
  float*       out   = (float*)d_out;
  const int B = in_sizes[2];

  // out = copy of W (updates are applied on top; inputs never mutated).
  (void)hipMemcpyAsync(out, W, (size_t)out_size * sizeof(float),
                       hipMemcpyDeviceToDevice, stream);

  const int wavesPerBlock = 8;              // 256 threads = 8 wave32 per WGP pass
  dim3 block(32 * wavesPerBlock);
  dim3 grid((unsigned)((B + wavesPerBlock - 1) / wavesPerBlock));
  verse_update_kernel<<<grid, block, 0, stream>>>(W, lut, u, v_pos, v_neg, out, B);
}